// KSpaceLoss_90821378441748
// MI455X (gfx1250) — compile-verified
//
#include <hip/hip_runtime.h>
#include <hip/hip_bf16.h>
#include <math.h>

// ---------------------------------------------------------------------------
// KSpaceLoss on MI455X (gfx1250)
//
// loss = 0.1*L1(ksp-ref)/L1(ref) + 0.1*L2(ksp-ref)/L2(ref)
// ksp[c,k,t] = mask[k] * sum_n exp(-2pi i (tx_k*gx + ty_k*gy)) * 0.5*img*smap
//
// Z==1  =>  E[k,n] = Ex[k,ix]*Ey[k,iy]  (separable phase, 48x fewer sincos)
// Stage 1: complex GEMM  Ey(8192x96) x Xmat(96x3072)  via v_wmma_f32_16x16x4_f32
//          (4 real WMMAs per K-step; A cached in VGPRs; B double-buffered in
//           LDS via global_load_async_to_lds_b128, waited with s_wait_asynccnt)
// Stage 2: per-k combine with Ex (fused, VALU) -> masked L1/L2 reduction
//
// 128-thread blocks (4 waves = 1 wave/SIMD) + __launch_bounds__(128,1) so the
// 96-VGPR A-operand cache stays in registers (round-2 version spilled).
// ---------------------------------------------------------------------------

#define NX   96
#define NY   96
#define NC   8
#define NT   4
#define NK   8192
#define NCOL (NX * 32)          // 3072 columns: col = ix*32 + (c*4 + t)
#define NPIX (NX * NY)          // 9216
#define KPB  32                 // k's per block

// dynamic LDS layout (bytes)
#define LDS_EY   0              // float2 Ey[32][96]            24 KB
#define LDS_EX   24576          // float2 Ex[32][96]            24 KB
#define LDS_BUF  49152          // 2 x { Xr[96][32], Xi[96][32] }  2 x 24 KB
#define BUF_BYTES 24576
#define LDS_TOTAL (LDS_BUF + 2 * BUF_BYTES)   // 98304 B = 96 KB

typedef float v2f __attribute__((ext_vector_type(2)));
typedef float v8f __attribute__((ext_vector_type(8)));

// ---------------- kernel 0: zero the 4 loss accumulators ----------------
__global__ void ksl_init_kernel(float* accum) {
    int i = threadIdx.x;
    if (i < 4) accum[i] = 0.0f;
}

// ---------------- kernel 1: build Xmat = 0.5 * img * smap ----------------
// Xr/Xi are 96(iy) x 3072(col) real planes, col = ix*32 + c*4 + t
__global__ void ksl_prep_kernel(const float2* __restrict__ img,
                                const float2* __restrict__ smaps,
                                float* __restrict__ Xr,
                                float* __restrict__ Xi) {
    int idx = blockIdx.x * blockDim.x + threadIdx.x;
    if (idx >= NY * NCOL) return;
    int iy  = idx / NCOL;
    int col = idx - iy * NCOL;
    int ix  = col >> 5;
    int ct  = col & 31;
    int c   = ct >> 2;
    int t   = ct & 3;
    float2 iv = img[(ix * NY + iy) * NT + t];          // (X,Y,1,1,T)
    float2 sv = smaps[c * NPIX + ix * NY + iy];        // (C,X,Y,1)
    Xr[idx] = 0.5f * (iv.x * sv.x - iv.y * sv.y);
    Xi[idx] = 0.5f * (iv.x * sv.y + iv.y * sv.x);
}

// async-stage one ix-block (96 x 32 cols, Re+Im) into LDS buffer `buf`.
// 1536 x b128 transfers per block, 12 per thread; ASYNCcnt-tracked.
__device__ __forceinline__ void ksl_async_fill(const float* __restrict__ Xr,
                                               const float* __restrict__ Xi,
                                               int ix, int buf, int tid) {
    const unsigned base = __builtin_amdgcn_groupstaticsize() + LDS_BUF +
                          (unsigned)buf * BUF_BYTES;
    #pragma unroll
    for (int u = 0; u < 6; ++u) {
        const int q   = tid + u * 128;        // 0..767 : 16B chunk id
        const int iy  = q >> 3;
        const int c16 = q & 7;
        const float* g0 = Xr + iy * NCOL + ix * 32 + c16 * 4;
        const float* g1 = Xi + iy * NCOL + ix * 32 + c16 * 4;
        const unsigned l0 = base + (unsigned)(iy * 32 + c16 * 4) * 4u;
        const unsigned l1 = l0 + 12288u;      // Xi plane
        asm volatile("global_load_async_to_lds_b128 %0, %1, off"
                     :: "v"(l0), "v"(g0) : "memory");
        asm volatile("global_load_async_to_lds_b128 %0, %1, off"
                     :: "v"(l1), "v"(g1) : "memory");
    }
}

// ---------------- kernel 2: fused NUFFT GEMM + loss ----------------
// grid: 256 blocks x 128 threads. Block b owns k in [b*32, b*32+32).
// Wave w (4 per block): ktile_local = w/2 (16 k's), ct-half = w%2 (16 cols).
__global__ void __launch_bounds__(128, 1)
ksl_main_kernel(const float* __restrict__ traj,
                const float* __restrict__ Xr,
                const float* __restrict__ Xi,
                const float2* __restrict__ kdata,
                const float* __restrict__ mask,
                float* __restrict__ accum) {
    extern __shared__ char ldsraw[];
    float2* Ey = (float2*)(ldsraw + LDS_EY);   // [32][96]
    float2* Ex = (float2*)(ldsraw + LDS_EX);   // [32][96]

    const int tid = threadIdx.x;
    const int k0  = blockIdx.x * KPB;

    // Build Ey (phase vs gy with ty) and Ex (phase vs gx with tx): cos, -sin
    for (int i = tid; i < 2 * KPB * NY; i += 128) {
        const bool isEx = (i >= KPB * NY);
        const int  j    = isEx ? i - KPB * NY : i;
        const int  kl   = j / NY;
        const int  g    = j - kl * NY;                     // ix or iy
        const float tr  = isEx ? traj[k0 + kl]             // tx row of (3,K)
                               : traj[NK + k0 + kl];       // ty row
        const float ph  = 6.28318530717958647692f * tr * (float)(g - NX / 2);
        float s, c;
        __sincosf(ph, &s, &c);
        ((float2*)ldsraw)[i] = make_float2(c, -s);         // exp(-i*ph)
    }
    // prefetch B block for ix=0 into buffer 0 while E tables settle
    ksl_async_fill(Xr, Xi, 0, 0, tid);
    asm volatile("s_wait_asynccnt 0x0" ::: "memory");
    __syncthreads();

    const int wave   = tid >> 5;
    const int lane   = tid & 31;
    const int ktl    = wave >> 1;          // 0..1
    const int cthalf = wave & 1;           // 0..1
    const int laneHi = lane >> 4;          // 0/1
    const int laneLo = lane & 15;
    const int klA    = ktl * 16 + laneLo;  // A-operand row (M = lane&15)
    const int cl     = cthalf * 16 + laneLo;   // column within 32-col block

    // cache the loop-invariant A operands (Ey row of this lane) in VGPRs:
    // A layout: K = j + 2*(lane>>4), j in {0,1}
    v2f Ar[24], Ai[24];
    #pragma unroll
    for (int s = 0; s < 24; ++s) {
        const int r0 = 4 * s + 2 * laneHi;
        const float2 e0 = Ey[klA * NY + r0];
        const float2 e1 = Ey[klA * NY + r0 + 1];
        Ar[s] = (v2f){e0.x, e1.x};
        Ai[s] = (v2f){e0.y, e1.y};
    }

    v8f a2r = {0, 0, 0, 0, 0, 0, 0, 0};    // stage-2 accumulators (16k x 16ct)
    v8f a2i = {0, 0, 0, 0, 0, 0, 0, 0};

    for (int ix = 0; ix < NX; ++ix) {
        // stream next B block while computing on the current one
        if (ix + 1 < NX) ksl_async_fill(Xr, Xi, ix + 1, (ix + 1) & 1, tid);

        const float* bb = (const float*)(ldsraw + LDS_BUF + (ix & 1) * BUF_BYTES) + cl;
        v8f cr = {0, 0, 0, 0, 0, 0, 0, 0};
        v8f ci = {0, 0, 0, 0, 0, 0, 0, 0};

        #pragma unroll
        for (int s = 0; s < 24; ++s) {
            const int r0 = 4 * s + 2 * laneHi;
            v2f Br = {bb[r0 * 32],        bb[(r0 + 1) * 32]};
            v2f Bi = {bb[3072 + r0 * 32], bb[3072 + (r0 + 1) * 32]};
            v2f mAi = {-Ai[s][0], -Ai[s][1]};
            // complex GEMM as 4 real WMMAs (f32 WMMA has no A/B NEG -> mAi)
            cr = __builtin_amdgcn_wmma_f32_16x16x4_f32(false, Ar[s], false, Br,
                                                       (short)0, cr, false, false);
            cr = __builtin_amdgcn_wmma_f32_16x16x4_f32(false, mAi,   false, Bi,
                                                       (short)0, cr, false, false);
            ci = __builtin_amdgcn_wmma_f32_16x16x4_f32(false, Ar[s], false, Bi,
                                                       (short)0, ci, false, false);
            ci = __builtin_amdgcn_wmma_f32_16x16x4_f32(false, Ai[s], false, Br,
                                                       (short)0, ci, false, false);
        }

        // stage 2: out[k,ct] += Ex[k,ix] * Btile[k,ct]; C row M = r + 8*(lane>>4)
        #pragma unroll
        for (int r = 0; r < 8; ++r) {
            const int ml = ktl * 16 + r + 8 * laneHi;
            const float2 ex = Ex[ml * NY + ix];
            a2r[r] += ex.x * cr[r] - ex.y * ci[r];
            a2i[r] += ex.x * ci[r] + ex.y * cr[r];
        }

        // own async writes for ix+1 done; all waves finished reading buf ix&1
        asm volatile("s_wait_asynccnt 0x0" ::: "memory");
        __syncthreads();
    }

    // per-element masked loss contributions
    float d1 = 0.f, a1 = 0.f, d2 = 0.f, a2 = 0.f;
    const int ct = cthalf * 16 + laneLo;
    const int c  = ct >> 2;
    const int t  = ct & 3;
    #pragma unroll
    for (int r = 0; r < 8; ++r) {
        const int k = k0 + ktl * 16 + r + 8 * laneHi;
        const float mk = mask[k];
        const float2 kd = kdata[(c * NK + k) * NT + t];    // (C,K,1,T)
        const float dr = mk * (a2r[r] - kd.x);
        const float di = mk * (a2i[r] - kd.y);
        const float rr = mk * kd.x, ri = mk * kd.y;
        d1 += sqrtf(dr * dr + di * di);
        a1 += sqrtf(rr * rr + ri * ri);
        d2 += dr * dr + di * di;
        a2 += rr * rr + ri * ri;
    }
    // wave32 reduction
    #pragma unroll
    for (int off = 16; off >= 1; off >>= 1) {
        d1 += __shfl_xor(d1, off);
        a1 += __shfl_xor(a1, off);
        d2 += __shfl_xor(d2, off);
        a2 += __shfl_xor(a2, off);
    }
    if (lane == 0) {
        unsafeAtomicAdd(&accum[0], d1);
        unsafeAtomicAdd(&accum[1], a1);
        unsafeAtomicAdd(&accum[2], d2);
        unsafeAtomicAdd(&accum[3], a2);
    }
}

// ---------------- kernel 3: finalize scalar ----------------
__global__ void ksl_final_kernel(const float* __restrict__ accum,
                                 float* __restrict__ out) {
    if (threadIdx.x == 0 && blockIdx.x == 0) {
        out[0] = 0.1f * (accum[0] / accum[1]) +
                 0.1f * (sqrtf(accum[2]) / sqrtf(accum[3]));
    }
}

extern "C" void kernel_launch(void* const* d_in, const int* in_sizes, int n_in,
                              void* d_out, int out_size, void* d_ws, size_t ws_size,
                              hipStream_t stream) {
    const float2* img   = (const float2*)d_in[0];  // images_reconstructed (complex64)
    const float*  traj  = (const float*)d_in[1];   // kspace_trajectory (3,K) f32
    const float2* kdata = (const float2*)d_in[2];  // kspace_data (complex64)
    const float*  mask  = (const float*)d_in[3];   // kspace_mask f32
    const float2* smaps = (const float2*)d_in[4];  // sensitivity_maps (complex64)
    float* out = (float*)d_out;

    // workspace layout: [0,16) accumulators | 256-aligned Xr plane | Xi plane
    float* accum = (float*)d_ws;
    float* Xr    = (float*)((char*)d_ws + 256);
    float* Xi    = Xr + NY * NCOL;

    ksl_init_kernel<<<1, 32, 0, stream>>>(accum);
    ksl_prep_kernel<<<(NY * NCOL + 255) / 256, 256, 0, stream>>>(img, smaps, Xr, Xi);
    // 256 blocks x 128 threads; 96 KB dynamic LDS (Ey + Ex + double B buffer)
    ksl_main_kernel<<<NK / KPB, 128, LDS_TOTAL, stream>>>(
        traj, Xr, Xi, kdata, mask, accum);
    ksl_final_kernel<<<1, 32, 0, stream>>>(accum, out);
}